// RealNVP_13245679141561
// MI455X (gfx1250) — compile-verified
//
#include <hip/hip_runtime.h>

typedef __attribute__((ext_vector_type(2))) float v2f;
typedef __attribute__((ext_vector_type(8))) float v8f;

#define FLOWS 16
#define NT 4            // 16-sample tiles carried per wave (weight-frag reuse)

// D = A(16x4) * B(4x16) + C, all f32. 8-arg VOP3P form per probe doc.
__device__ __forceinline__ v8f wmma4(v2f a, v2f b, v8f c) {
  return __builtin_amdgcn_wmma_f32_16x16x4_f32(false, a, false, b, (short)0, c,
                                               false, false);
}

__device__ __forceinline__ float fast_tanh(float x) {
  // monotone, no NaN at +/-inf: e->inf => 1, e->0 => -1
  float e = __expf(2.0f * x);
  return 1.0f - 2.0f / (e + 1.0f);
}

// LDS layout (floats). Weights zero-padded to WMMA shapes. Total 15936*4 = 63.7KB.
#define OFF_W1 0        // [16][12]
#define OFF_B1 192      // [16][12]
#define OFF_W2 384      // [16][12][16]
#define OFF_B2 3456     // [16][16]
#define OFF_W3 3712     // [16][12][20]
#define OFF_B3 7552     // [16][20]
#define OFF_W4 7872     // [16][20][16]
#define OFF_B4 12992    // [16][16]
#define OFF_SCR 13248   // [8 waves][16][21]
#define LDS_TOT 15936

__launch_bounds__(256)
__global__ void realnvp_wmma_kernel(
    const float* __restrict__ in,
    const float* __restrict__ W1, const float* __restrict__ b1,
    const float* __restrict__ W2, const float* __restrict__ b2,
    const float* __restrict__ W3, const float* __restrict__ b3,
    const float* __restrict__ W4, const float* __restrict__ b4,
    float* __restrict__ out, int B, int nGroups)
{
  __shared__ float smem[LDS_TOT];
  const int tid = threadIdx.x;

  // ---- stage weights into LDS, zero-padded ----
  for (int i = tid; i < OFF_SCR; i += 256) smem[i] = 0.0f;
  __syncthreads();
  float* sW1 = smem + OFF_W1;  float* sB1 = smem + OFF_B1;
  float* sW2 = smem + OFF_W2;  float* sB2 = smem + OFF_B2;
  float* sW3 = smem + OFF_W3;  float* sB3 = smem + OFF_B3;
  float* sW4 = smem + OFF_W4;  float* sB4 = smem + OFF_B4;
  for (int i = tid; i < 160; i += 256) {                 // W1 (16,1,10), b1 (16,10)
    int f = i / 10, k = i % 10;
    sW1[f * 12 + k] = W1[i];  sB1[f * 12 + k] = b1[i];
  }
  for (int i = tid; i < 1600; i += 256) {                // W2 (16,10,10)
    int f = i / 100, k = (i / 10) % 10, n = i % 10;
    sW2[(f * 12 + k) * 16 + n] = W2[i];
  }
  for (int i = tid; i < 160; i += 256) {                 // b2 (16,10)
    int f = i / 10, n = i % 10;
    sB2[f * 16 + n] = b2[i];
  }
  for (int i = tid; i < 3200; i += 256) {                // W3 (16,10,20)
    int f = i / 200, k = (i / 20) % 10, n = i % 20;
    sW3[(f * 12 + k) * 20 + n] = W3[i];
  }
  for (int i = tid; i < 320; i += 256) sB3[i] = b3[i];   // b3 (16,20), same stride
  for (int i = tid; i < 640; i += 256) {                 // W4 (16,20,2)
    int f = i / 40, r = i % 40, k = r / 2, n = r % 2;
    sW4[(f * 20 + k) * 16 + n] = W4[i];
  }
  for (int i = tid; i < 32; i += 256) {                  // b4 (16,2)
    int f = i / 2, n = i % 2;
    sB4[f * 16 + n] = b4[i];
  }
  __syncthreads();

  const int lane = tid & 31;
  const int lm   = lane & 15;   // row (A/state) or column (B/D) index
  const int hl   = lane >> 4;   // lane half -> K sub-pair selector
  const int wv   = tid >> 5;
  float* scr = smem + OFF_SCR + wv * 336;   // per-wave [16 rows][stride 21]

  const int waveGlobal = blockIdx.x * 8 + wv;
  const int nWaves = gridDim.x * 8;
  const v8f vzero = {0.f, 0.f, 0.f, 0.f, 0.f, 0.f, 0.f, 0.f};

  for (int grp = waveGlobal; grp < nGroups; grp += nWaves) {
    const int base = grp * (NT * 16);
    float z0s[NT], z1s[NT], ldet[NT];
#pragma unroll
    for (int t = 0; t < NT; ++t) {
      int s = base + t * 16 + lm;
      s = (s < B) ? s : (B - 1);
      float2 zz = ((const float2*)in)[s];     // (B,2) row-major
      z0s[t] = zz.x;  z1s[t] = zz.y;  ldet[t] = 0.0f;
    }

#pragma unroll 1
    for (int f = 0; f < FLOWS; ++f) {
      // ---- per-flow fragments (reused across NT tiles) ----
      const float* pw1 = sW1 + f * 12 + 2 * hl;   // 8B aligned
      const float* pb1 = sB1 + f * 12 + 2 * hl;
      v2f w1c[3], b1c[3], b2f[3], b3f0[3], b3f1[3], b4f[5];
      const int col1 = 16 + (lm & 3);  // valid lanes lm<4 (cols 16..19); rest unused
#pragma unroll
      for (int c = 0; c < 3; ++c) {
        w1c[c] = *(const v2f*)(pw1 + 4 * c);
        b1c[c] = *(const v2f*)(pb1 + 4 * c);
        const int k = 4 * c + 2 * hl;
        b2f[c][0] = sW2[(f * 12 + k) * 16 + lm];
        b2f[c][1] = sW2[(f * 12 + k + 1) * 16 + lm];
        b3f0[c][0] = sW3[(f * 12 + k) * 20 + lm];
        b3f0[c][1] = sW3[(f * 12 + k + 1) * 20 + lm];
        b3f1[c][0] = sW3[(f * 12 + k) * 20 + col1];
        b3f1[c][1] = sW3[(f * 12 + k + 1) * 20 + col1];
      }
#pragma unroll
      for (int c = 0; c < 5; ++c) {
        const int k = 4 * c + 2 * hl;
        b4f[c][0] = sW4[(f * 20 + k) * 16 + lm];
        b4f[c][1] = sW4[(f * 20 + k + 1) * 16 + lm];
      }
      const float bias2  = sB2[f * 16 + lm];
      const float bias30 = sB3[f * 20 + lm];
      const float bias31 = sB3[f * 20 + col1];
      const float bias4  = sB4[f * 16 + lm];
      const int rowb = 8 * hl;

#pragma unroll
      for (int t = 0; t < NT; ++t) {
        // ---- layer1 (built in A layout) + layer2 WMMA ----
        v8f acc = vzero;
#pragma unroll
        for (int c = 0; c < 3; ++c) {
          v2f a;
          a[0] = fmaxf(fmaf(z0s[t], w1c[c][0], b1c[c][0]), 0.0f);
          a[1] = fmaxf(fmaf(z0s[t], w1c[c][1], b1c[c][1]), 0.0f);
          acc = wmma4(a, b2f[c], acc);
        }
        // bias+relu in D layout; pad cols (n>=10) are exact zeros -> store all 16
#pragma unroll
        for (int v = 0; v < 8; ++v)
          scr[(v + rowb) * 21 + lm] = fmaxf(acc[v] + bias2, 0.0f);
        asm volatile("" ::: "memory");   // same-wave DS is in-order; pin compiler order

        // ---- layer3: D->A via LDS transpose, two N tiles ----
        v2f a3[3];
#pragma unroll
        for (int c = 0; c < 3; ++c) {
          const int k = 4 * c + 2 * hl;
          a3[c][0] = scr[lm * 21 + k];
          a3[c][1] = scr[lm * 21 + k + 1];
        }
        asm volatile("" ::: "memory");
        v8f acc3a = vzero, acc3b = vzero;
#pragma unroll
        for (int c = 0; c < 3; ++c) acc3a = wmma4(a3[c], b3f0[c], acc3a);
#pragma unroll
        for (int c = 0; c < 3; ++c) acc3b = wmma4(a3[c], b3f1[c], acc3b);
#pragma unroll
        for (int v = 0; v < 8; ++v)
          scr[(v + rowb) * 21 + lm] = fast_tanh(acc3a[v] + bias30);
        if (lm < 4) {
#pragma unroll
          for (int v = 0; v < 8; ++v)
            scr[(v + rowb) * 21 + 16 + lm] = fast_tanh(acc3b[v] + bias31);
        }
        asm volatile("" ::: "memory");

        // ---- layer4: K=20 exactly (5 chunks), only cols 0,1 of D used ----
        v8f acc4 = vzero;
#pragma unroll
        for (int c = 0; c < 5; ++c) {
          const int k = 4 * c + 2 * hl;
          v2f a4;
          a4[0] = scr[lm * 21 + k];
          a4[1] = scr[lm * 21 + k + 1];
          acc4 = wmma4(a4, b4f[c], acc4);
        }
        asm volatile("" ::: "memory");
        if (lm < 2) {
#pragma unroll
          for (int v = 0; v < 8; ++v)
            scr[(v + rowb) * 21 + lm] = acc4[v] + bias4;
        }
        asm volatile("" ::: "memory");
        const float ls = scr[lm * 21 + 0];
        const float sh = scr[lm * 21 + 1];

        // coupling update + swap (matches the scan carry)
        const float z1n = __expf(ls) * z1s[t] + sh;
        ldet[t] += ls;
        z1s[t] = z0s[t];
        z0s[t] = z1n;
      }
    }

    // ---- finalize: output z0_new = carry z1, z1_new = carry z0 ----
#pragma unroll
    for (int t = 0; t < NT; ++t) {
      const int s = base + t * 16 + lm;
      if (hl == 0 && s < B) {
        const float zn0 = z1s[t], zn1 = z0s[t];
        const float lp = -0.5f * (zn0 * zn0 + zn1 * zn1)
                         - 1.8378770664093453f + ldet[t];
        out[s] = lp;                       // output 0: (B,1)
        float2 o; o.x = zn0; o.y = zn1;
        ((float2*)(out + B))[s] = o;       // output 1: (B,2)
      }
    }
  }
}

extern "C" void kernel_launch(void* const* d_in, const int* in_sizes, int n_in,
                              void* d_out, int out_size, void* d_ws, size_t ws_size,
                              hipStream_t stream) {
  const float* in = (const float*)d_in[0];
  const float* W1 = (const float*)d_in[1];
  const float* b1 = (const float*)d_in[2];
  const float* W2 = (const float*)d_in[3];
  const float* b2 = (const float*)d_in[4];
  const float* W3 = (const float*)d_in[5];
  const float* b3 = (const float*)d_in[6];
  const float* W4 = (const float*)d_in[7];
  const float* b4 = (const float*)d_in[8];
  float* out = (float*)d_out;

  const int B = in_sizes[0] / 2;                    // samples
  const int tiles = (B + 15) / 16;
  const int nGroups = (tiles + NT - 1) / NT;
  int blocks = (nGroups + 7) / 8;                   // 8 waves per 256-thread block
  if (blocks > 16384) blocks = 16384;
  if (blocks < 1) blocks = 1;

  realnvp_wmma_kernel<<<blocks, 256, 0, stream>>>(in, W1, b1, W2, b2, W3, b3,
                                                  W4, b4, out, B, nGroups);
}